// GCNWithNormAndDropout_66245575574018
// MI455X (gfx1250) — compile-verified
//
#include <hip/hip_runtime.h>
#include <hip/hip_bf16.h>
#include <math.h>

typedef float v8f __attribute__((ext_vector_type(8)));
typedef float v2f __attribute__((ext_vector_type(2)));

#define D_HID 128
#define D_OUT 64
#define BN_EPS 1e-5f

// ---------------- degree / normalization ----------------

__global__ void init_deg_stats(float* deg, float* stats, int N) {
    int i = blockIdx.x * blockDim.x + threadIdx.x;
    int stride = blockDim.x * gridDim.x;
    for (; i < N; i += stride) {
        deg[i] = 1.0f;               // self loop
        if (i < 256) stats[i] = 0.0f; // bn sum / sumsq
    }
}

__global__ void count_edges(const int* __restrict__ col, float* deg, int E) {
    int e = blockIdx.x * blockDim.x + threadIdx.x;
    if (e < E) atomicAdd(&deg[col[e]], 1.0f);
}

__global__ void deg_to_dis(float* deg, int N) {
    int i = blockIdx.x * blockDim.x + threadIdx.x;
    if (i < N) deg[i] = rsqrtf(deg[i]);
}

// ---------------- GEMM1: h1 = x @ W1 + b1 (f32 WMMA) ----------------
// grid.x = N/16 row tiles; 8 waves per block = 8 column tiles (128 cols)

__global__ void gemm1_wmma(const float* __restrict__ x, const float* __restrict__ W1,
                           const float* __restrict__ b1, float* __restrict__ h1, int N) {
    const int wave  = threadIdx.x >> 5;      // 0..7 -> col tile
    const int lane  = threadIdx.x & 31;
    const int row0  = blockIdx.x * 16;
    if (row0 >= N) return;
    const int col0  = wave * 16;
    const int m     = lane & 15;
    const int n     = lane & 15;
    const int khalf = lane >> 4;             // 0 or 1

    v8f c;
    const float bias = b1[col0 + n];
    #pragma unroll
    for (int r = 0; r < 8; ++r) c[r] = bias;

    const float* arow = x + (size_t)(row0 + m) * D_HID + 2 * khalf;
    #pragma unroll 4
    for (int k = 0; k < 128; k += 4) {
        v2f a, b;
        a.x = arow[k];
        a.y = arow[k + 1];
        const int kb = k + 2 * khalf;
        b.x = W1[(size_t)(kb + 0) * D_HID + col0 + n];
        b.y = W1[(size_t)(kb + 1) * D_HID + col0 + n];
        c = __builtin_amdgcn_wmma_f32_16x16x4_f32(false, a, false, b, (short)0, c, false, false);
    }

    const int mbase = khalf * 8;
    #pragma unroll
    for (int r = 0; r < 8; ++r)
        h1[(size_t)(row0 + mbase + r) * D_HID + col0 + n] = c[r];
}

// ---------------- propagate 1 (128 channels) ----------------

__global__ void selfloop1(const float* __restrict__ h1, const float* __restrict__ dis,
                          float* __restrict__ hagg, int total) {
    int i = blockIdx.x * blockDim.x + threadIdx.x;
    if (i < total) {
        float d = dis[i >> 7];               // /128
        hagg[i] = d * d * h1[i];
    }
}

// one wave per edge; lane handles 4 channels (float4 load, 4 atomic adds)
__global__ void scatter1(const int* __restrict__ row, const int* __restrict__ col,
                         const float* __restrict__ dis, const float* __restrict__ h1,
                         float* hagg, int E) {
    const int wave = threadIdx.x >> 5;
    const int lane = threadIdx.x & 31;
    const int e = blockIdx.x * 8 + wave;
    if (e >= E) return;
    const int r = row[e];
    const int c = col[e];
    const float w = dis[r] * dis[c];
    const int base = lane * 4;
    const float4 v = *(const float4*)(h1 + (size_t)r * D_HID + base);
    float* dst = hagg + (size_t)c * D_HID + base;
    atomicAdd(dst + 0, w * v.x);
    atomicAdd(dst + 1, w * v.y);
    atomicAdd(dst + 2, w * v.z);
    atomicAdd(dst + 3, w * v.w);
}

// ---------------- BatchNorm stats ----------------

__global__ void bn_stats(const float* __restrict__ hagg, float* bsum, float* bsq, int total) {
    __shared__ float ssum[D_HID];
    __shared__ float ssq[D_HID];
    for (int i = threadIdx.x; i < D_HID; i += blockDim.x) { ssum[i] = 0.0f; ssq[i] = 0.0f; }
    __syncthreads();
    int stride = blockDim.x * gridDim.x;
    for (int i = blockIdx.x * blockDim.x + threadIdx.x; i < total; i += stride) {
        float v = hagg[i];
        atomicAdd(&ssum[i & 127], v);
        atomicAdd(&ssq[i & 127], v * v);
    }
    __syncthreads();
    for (int i = threadIdx.x; i < D_HID; i += blockDim.x) {
        atomicAdd(&bsum[i], ssum[i]);
        atomicAdd(&bsq[i], ssq[i]);
    }
}

__global__ void bn_finalize(const float* __restrict__ bsum, const float* __restrict__ bsq,
                            const float* __restrict__ gamma, const float* __restrict__ beta,
                            float* scale, float* shift, int N) {
    int c = threadIdx.x;
    if (c < D_HID) {
        float inv = 1.0f / (float)N;
        float mu = bsum[c] * inv;
        float var = bsq[c] * inv - mu * mu;
        float s = gamma[c] * rsqrtf(var + BN_EPS);
        scale[c] = s;
        shift[c] = beta[c] - mu * s;
    }
}

// ---------------- GEMM2: h2 = relu(bn(h1agg)) @ W2 + b2 (f32 WMMA) ----------------
// 8 waves per block: 2 row tiles x 4 col tiles (64 cols)

__global__ void gemm2_wmma(const float* __restrict__ hagg, const float* __restrict__ scale,
                           const float* __restrict__ shift, const float* __restrict__ W2,
                           const float* __restrict__ b2, float* __restrict__ h2, int N) {
    const int wave  = threadIdx.x >> 5;
    const int lane  = threadIdx.x & 31;
    const int row0  = (blockIdx.x * 2 + (wave >> 2)) * 16;
    if (row0 >= N) return;
    const int col0  = (wave & 3) * 16;
    const int m     = lane & 15;
    const int n     = lane & 15;
    const int khalf = lane >> 4;

    v8f c;
    const float bias = b2[col0 + n];
    #pragma unroll
    for (int r = 0; r < 8; ++r) c[r] = bias;

    const float* arow = hagg + (size_t)(row0 + m) * D_HID;
    #pragma unroll 4
    for (int k = 0; k < 128; k += 4) {
        const int ka = k + 2 * khalf;
        v2f a, b;
        a.x = fmaxf(arow[ka + 0] * scale[ka + 0] + shift[ka + 0], 0.0f);
        a.y = fmaxf(arow[ka + 1] * scale[ka + 1] + shift[ka + 1], 0.0f);
        b.x = W2[(size_t)(ka + 0) * D_OUT + col0 + n];
        b.y = W2[(size_t)(ka + 1) * D_OUT + col0 + n];
        c = __builtin_amdgcn_wmma_f32_16x16x4_f32(false, a, false, b, (short)0, c, false, false);
    }

    const int mbase = khalf * 8;
    #pragma unroll
    for (int r = 0; r < 8; ++r)
        h2[(size_t)(row0 + mbase + r) * D_OUT + col0 + n] = c[r];
}

// ---------------- propagate 2 (64 channels) into d_out ----------------

__global__ void selfloop2(const float* __restrict__ h2, const float* __restrict__ dis,
                          float* __restrict__ out, int total) {
    int i = blockIdx.x * blockDim.x + threadIdx.x;
    if (i < total) {
        float d = dis[i >> 6];               // /64
        out[i] = d * d * h2[i];
    }
}

// one wave per edge; lane handles 2 channels
__global__ void scatter2(const int* __restrict__ row, const int* __restrict__ col,
                         const float* __restrict__ dis, const float* __restrict__ h2,
                         float* out, int E) {
    const int wave = threadIdx.x >> 5;
    const int lane = threadIdx.x & 31;
    const int e = blockIdx.x * 8 + wave;
    if (e >= E) return;
    const int r = row[e];
    const int c = col[e];
    const float w = dis[r] * dis[c];
    const int base = lane * 2;
    const float2 v = *(const float2*)(h2 + (size_t)r * D_OUT + base);
    float* dst = out + (size_t)c * D_OUT + base;
    atomicAdd(dst + 0, w * v.x);
    atomicAdd(dst + 1, w * v.y);
}

// ---------------- row-wise log_softmax, in place (wave per row) ----------------

__global__ void logsoftmax_rows(float* out, int N) {
    const int wave = threadIdx.x >> 5;
    const int lane = threadIdx.x & 31;
    const int rowi = blockIdx.x * 8 + wave;
    if (rowi >= N) return;
    float* p = out + (size_t)rowi * D_OUT;
    float v0 = p[lane];
    float v1 = p[lane + 32];
    float mx = fmaxf(v0, v1);
    #pragma unroll
    for (int off = 16; off >= 1; off >>= 1)
        mx = fmaxf(mx, __shfl_xor(mx, off, 32));
    float s = __expf(v0 - mx) + __expf(v1 - mx);
    #pragma unroll
    for (int off = 16; off >= 1; off >>= 1)
        s += __shfl_xor(s, off, 32);
    float l = mx + __logf(s);
    p[lane]      = v0 - l;
    p[lane + 32] = v1 - l;
}

// ---------------- host launcher ----------------

extern "C" void kernel_launch(void* const* d_in, const int* in_sizes, int n_in,
                              void* d_out, int out_size, void* d_ws, size_t ws_size,
                              hipStream_t stream) {
    const float* x     = (const float*)d_in[0];
    const float* W1    = (const float*)d_in[1];
    const float* b1    = (const float*)d_in[2];
    const float* gamma = (const float*)d_in[3];
    const float* beta  = (const float*)d_in[4];
    const float* W2    = (const float*)d_in[5];
    const float* b2    = (const float*)d_in[6];
    const int*   ei    = (const int*)d_in[7];

    const int N = in_sizes[0] / D_HID;
    const int E = in_sizes[7] / 2;
    const int* erow = ei;
    const int* ecol = ei + E;

    float* ws    = (float*)d_ws;
    float* dis   = ws;                                   // N
    float* h1    = dis + N;                              // N*128 (reused as h2)
    float* hagg  = h1 + (size_t)N * D_HID;               // N*128
    float* stats = hagg + (size_t)N * D_HID;             // 512: sum,sumsq,scale,shift
    float* bsum  = stats;
    float* bsq   = stats + 128;
    float* scal  = stats + 256;
    float* shft  = stats + 384;
    float* h2    = h1;                                   // reuse (h1 dead after propagate1)
    float* outp  = (float*)d_out;

    // degree / normalization
    init_deg_stats<<<(N + 255) / 256, 256, 0, stream>>>(dis, stats, N);
    count_edges<<<(E + 255) / 256, 256, 0, stream>>>(ecol, dis, E);
    deg_to_dis<<<(N + 255) / 256, 256, 0, stream>>>(dis, N);

    // layer 1
    gemm1_wmma<<<(N + 15) / 16, 256, 0, stream>>>(x, W1, b1, h1, N);
    selfloop1<<<((size_t)N * D_HID + 255) / 256, 256, 0, stream>>>(h1, dis, hagg, N * D_HID);
    scatter1<<<(E + 7) / 8, 256, 0, stream>>>(erow, ecol, dis, h1, hagg, E);

    // batchnorm stats
    bn_stats<<<1024, 256, 0, stream>>>(hagg, bsum, bsq, N * D_HID);
    bn_finalize<<<1, 128, 0, stream>>>(bsum, bsq, gamma, beta, scal, shft, N);

    // layer 2 (BN+ReLU fused into A-operand load)
    gemm2_wmma<<<(N + 31) / 32, 256, 0, stream>>>(hagg, scal, shft, W2, b2, h2, N);
    selfloop2<<<((size_t)N * D_OUT + 255) / 256, 256, 0, stream>>>(h2, dis, outp, N * D_OUT);
    scatter2<<<(E + 7) / 8, 256, 0, stream>>>(erow, ecol, dis, h2, outp, E);

    // log_softmax in place
    logsoftmax_rows<<<(N + 7) / 8, 256, 0, stream>>>(outp, N);
}